// DeformableAttentionV2_74904229642680
// MI455X (gfx1250) — compile-verified
//
#include <hip/hip_runtime.h>
#include <hip/hip_bf16.h>
#include <math.h>

typedef float v2f __attribute__((ext_vector_type(2)));
typedef float v8f __attribute__((ext_vector_type(8)));

#define CDIM   256        // embed dim
#define NDIM   36         // MAX_NL*NH*NP*2
#define NLVL   2
#define NPTS   9
#define NPF    (NLVL*NPTS)   // 18 flattened sample points
#define KSTEPS (CDIM/4)      // 64 K-steps of K=4 for f32 WMMA

// ---------------------------------------------------------------------------
// Kernel 1: off[nq,36] = query[nq,256] @ W_off[256,36] + b_off
// f32 WMMA 16x16x4, one 16-row M-tile per wave, 3 N-tiles (48 cols, 36 used).
// B fragments pre-packed into LDS in exact WMMA lane order:
//   lane l: half = l>>4, idx = l&15; B frag pair = W[k+2*half][n0+idx],
//   W[k+2*half+1][n0+idx]  (mirrors documented 16x4 f32 A layout).
// ---------------------------------------------------------------------------
__global__ __launch_bounds__(256) void offsets_gemm_wmma(
        const float* __restrict__ query,
        const float* __restrict__ Woff,
        const float* __restrict__ boff,
        float* __restrict__ off,
        int nq, int ntiles) {
    __shared__ float wpack[KSTEPS * 3 * 32 * 2];   // 48 KB, fragment-ordered B

    // Pack W_off (36 KB, L2-hot) into fragment order once per block.
    for (int e = threadIdx.x; e < KSTEPS * 3 * 32; e += blockDim.x) {
        int ks = e / 96, r = e % 96;
        int t = r / 32, l = r % 32;
        int half = l >> 4, idx = l & 15;
        int col = t * 16 + idx;
        int k = ks * 4 + half * 2;
        float b0 = 0.f, b1 = 0.f;
        if (col < NDIM) {
            b0 = Woff[k * NDIM + col];
            b1 = Woff[(k + 1) * NDIM + col];
        }
        wpack[e * 2 + 0] = b0;
        wpack[e * 2 + 1] = b1;
    }
    __syncthreads();

    int wave = threadIdx.x >> 5;
    int lane = threadIdx.x & 31;
    int tile = blockIdx.x * (blockDim.x >> 5) + wave;
    if (tile >= ntiles) return;                 // wave-uniform: EXEC stays all-1s

    int m0 = tile * 16;
    int half = lane >> 4, idx = lane & 15;
    int arow = m0 + idx; if (arow > nq - 1) arow = nq - 1;
    const float* aptr = query + (size_t)arow * CDIM + half * 2;

    v8f c0 = {}, c1 = {}, c2 = {};
    for (int ks = 0; ks < KSTEPS; ++ks) {
        v2f a  = *(const v2f*)(aptr + ks * 4);                          // 8B-aligned
        v2f b0 = *(const v2f*)(&wpack[((ks * 3 + 0) * 32 + lane) * 2]);
        v2f b1 = *(const v2f*)(&wpack[((ks * 3 + 1) * 32 + lane) * 2]);
        v2f b2 = *(const v2f*)(&wpack[((ks * 3 + 2) * 32 + lane) * 2]);
        c0 = __builtin_amdgcn_wmma_f32_16x16x4_f32(false, a, false, b0, (short)0, c0, false, false);
        c1 = __builtin_amdgcn_wmma_f32_16x16x4_f32(false, a, false, b1, (short)0, c1, false, false);
        c2 = __builtin_amdgcn_wmma_f32_16x16x4_f32(false, a, false, b2, (short)0, c2, false, false);
    }

    // D layout: lane<16 -> (M=r, N=idx); lane>=16 -> (M=r+8, N=idx). Add bias, store.
    for (int t = 0; t < 3; ++t) {
        int col = t * 16 + idx;
        if (col >= NDIM) continue;
        float bias = boff[col];
        v8f c = (t == 0) ? c0 : ((t == 1) ? c1 : c2);
        for (int r = 0; r < 8; ++r) {
            int row = m0 + r + half * 8;
            if (row < nq) off[(size_t)row * NDIM + col] = c[r] + bias;
        }
    }
}

// ---------------------------------------------------------------------------
// Kernel 2: bilinear gather + attention. 1 block per query, 256 thr = channels.
// value is L2-resident (33.5 MB << 192 MB); each corner read is a contiguous
// 1 KB row, coalesced across the 256 channel-threads.
// ---------------------------------------------------------------------------
__global__ __launch_bounds__(CDIM) void deform_attn_sample(
        const float* __restrict__ query,
        const float* __restrict__ value,
        const float* __restrict__ refpts,
        const int*   __restrict__ sshapes,
        const float* __restrict__ off,
        float* __restrict__ out,
        int nq, int nv) {
    __shared__ float kv_s[NPF][CDIM + 1];   // +1 pad: avoid bank conflicts
    __shared__ float q_s[CDIM];
    __shared__ float off_s[NDIM];
    __shared__ float ref_s[4];
    __shared__ float hw_s[4];               // H0,W0,H1,W1 as float
    __shared__ float sc_s[NPF];

    int q = blockIdx.x;
    int t = threadIdx.x;
    float qv = query[(size_t)q * CDIM + t];
    q_s[t] = qv;
    if (t < NDIM)                 off_s[t]        = off[(size_t)q * NDIM + t];
    if (t >= NDIM && t < NDIM + 4) ref_s[t - NDIM] = refpts[(size_t)q * 4 + (t - NDIM)];
    if (t >= 64 && t < 68)        hw_s[t - 64]    = (float)sshapes[t - 64];
    __syncthreads();

    // kv[p][c] via bilinear sampling (zeros padding, align_corners=False)
    for (int p = 0; p < NPF; ++p) {
        int l = p / NPTS;
        float Hf = hw_s[l * 2 + 0], Wf = hw_s[l * 2 + 1];
        float locx = ref_s[l * 2 + 0] + off_s[p * 2 + 0] / Wf;
        float locy = ref_s[l * 2 + 1] + off_s[p * 2 + 1] / Hf;
        float x = locx * Wf - 0.5f;         // == (grid_x+1)*W/2 - 0.5
        float y = locy * Hf - 0.5f;
        float x0 = floorf(x), y0 = floorf(y);
        float wx1 = x - x0, wy1 = y - y0;
        float wx0 = 1.f - wx1, wy0 = 1.f - wy1;
        float x1 = x0 + 1.f, y1 = y0 + 1.f;
        int W = (int)Wf;
        int xi0 = (int)fminf(fmaxf(x0, 0.f), Wf - 1.f);
        int xi1 = (int)fminf(fmaxf(x1, 0.f), Wf - 1.f);
        int yi0 = (int)fminf(fmaxf(y0, 0.f), Hf - 1.f);
        int yi1 = (int)fminf(fmaxf(y1, 0.f), Hf - 1.f);
        float vx0 = (x0 >= 0.f && x0 <= Wf - 1.f) ? 1.f : 0.f;
        float vx1 = (x1 >= 0.f && x1 <= Wf - 1.f) ? 1.f : 0.f;
        float vy0 = (y0 >= 0.f && y0 <= Hf - 1.f) ? 1.f : 0.f;
        float vy1 = (y1 >= 0.f && y1 <= Hf - 1.f) ? 1.f : 0.f;
        const float* vb = value + (size_t)l * nv * CDIM;
        float acc;
        acc  = vb[(size_t)(yi0 * W + xi0) * CDIM + t] * (wx0 * wy0 * vx0 * vy0);
        acc += vb[(size_t)(yi0 * W + xi1) * CDIM + t] * (wx1 * wy0 * vx1 * vy0);
        acc += vb[(size_t)(yi1 * W + xi0) * CDIM + t] * (wx0 * wy1 * vx0 * vy1);
        acc += vb[(size_t)(yi1 * W + xi1) * CDIM + t] * (wx1 * wy1 * vx1 * vy1);
        kv_s[p][t] = acc;
    }
    __syncthreads();

    // scores[p] = (q . kv_p) * sqrt(256); wave32 shuffle reduction, 8 waves / 18 p
    int wave = t >> 5, lane = t & 31;
    for (int p = wave; p < NPF; p += (CDIM / 32)) {
        float part = 0.f;
        for (int i = 0; i < CDIM / 32; ++i) {
            int c = lane + i * 32;
            part += q_s[c] * kv_s[p][c];
        }
        for (int m = 16; m > 0; m >>= 1) part += __shfl_xor(part, m, 32);
        if (lane == 0) sc_s[p] = part * 16.0f;   // sqrt(256)
    }
    __syncthreads();

    // softmax over 18 (redundant per thread, trivial) + weighted sum back
    float mx = -1e30f;
    for (int p = 0; p < NPF; ++p) mx = fmaxf(mx, sc_s[p]);
    float e[NPF], sum = 0.f;
    for (int p = 0; p < NPF; ++p) { e[p] = __expf(sc_s[p] - mx); sum += e[p]; }
    float inv = 1.f / sum;
    float o = 0.f;
    for (int p = 0; p < NPF; ++p) o += e[p] * kv_s[p][t];
    out[(size_t)q * CDIM + t] = o * inv;
}

// ---------------------------------------------------------------------------
extern "C" void kernel_launch(void* const* d_in, const int* in_sizes, int n_in,
                              void* d_out, int out_size, void* d_ws, size_t ws_size,
                              hipStream_t stream) {
    const float* query = (const float*)d_in[0];
    /* d_in[1] = key, unused by forward */
    const float* value = (const float*)d_in[2];
    const float* refp  = (const float*)d_in[3];
    const int*   ss    = (const int*)d_in[4];
    const float* Woff  = (const float*)d_in[5];
    const float* boff  = (const float*)d_in[6];
    float* out = (float*)d_out;

    int nq = in_sizes[0] / CDIM;            // 20000
    int nv = in_sizes[2] / (NLVL * CDIM);   // 16384

    float* off = (float*)d_ws;              // nq*36 floats of scratch

    int ntiles  = (nq + 15) / 16;
    int blocks1 = (ntiles + 7) / 8;         // 8 waves (M-tiles) per 256-thr block
    offsets_gemm_wmma<<<blocks1, 256, 0, stream>>>(query, Woff, boff, off, nq, ntiles);
    deform_attn_sample<<<nq, CDIM, 0, stream>>>(query, value, refp, ss, off, out, nq, nv);
}